// HAN_9552007266961
// MI455X (gfx1250) — compile-verified
//
#include <hip/hip_runtime.h>
#include <hip/hip_bf16.h>

// ---------------------------------------------------------------------------
// HAN forward for MI455X (gfx1250, wave32, WMMA).
// N=100000 nodes, E=1.6M edges/metapath, DIN=DH=128, H=8 heads, D=16, DOUT=64.
// GEMMs use V_WMMA_F32_16X16X4_F32 (fp32 semantics; GEMM cost is negligible
// vs the L2-resident edge scatter phase, so no reason to drop precision).
// ---------------------------------------------------------------------------

typedef __attribute__((ext_vector_type(2))) float v2f;
typedef __attribute__((ext_vector_type(8))) float v8f;

#define NVAL   100000
#define DHVAL  128
#define HVAL   8
#define DVAL   16
#define DOUTVAL 64
#define NEG_SLOPE 0.2f
#define ENC_NEG_INF 0x007FFFFFu   // monotonic encoding of -inf

// Order-preserving float -> uint map (for deterministic atomic max).
__device__ __forceinline__ unsigned encf(float f) {
    unsigned u = __float_as_uint(f);
    return (u & 0x80000000u) ? ~u : (u | 0x80000000u);
}
__device__ __forceinline__ float decf(unsigned e) {
    return (e & 0x80000000u) ? __uint_as_float(e & 0x7FFFFFFFu)
                             : __uint_as_float(~e);
}

// ---------------------------------------------------------------------------
// 0) Workspace init: m0/m1 = enc(-inf), s0/s1 = 0, out0/out1 = 0, score = 0
// ---------------------------------------------------------------------------
__global__ void han_init_kernel(unsigned* __restrict__ m0, unsigned* __restrict__ m1,
                                float* __restrict__ s0, float* __restrict__ s1,
                                float* __restrict__ out0, float* __restrict__ out1,
                                float* __restrict__ score, int nh, int ndh) {
    int i = blockIdx.x * blockDim.x + threadIdx.x;
    if (i < nh) { m0[i] = ENC_NEG_INF; m1[i] = ENC_NEG_INF; s0[i] = 0.f; s1[i] = 0.f; }
    if (i < ndh) { out0[i] = 0.f; out1[i] = 0.f; }
    if (i < 2) score[i] = 0.f;
}

// ---------------------------------------------------------------------------
// 1) Projection: h[n,:] = x[n,:] @ proj_w + proj_b   (N x 128 @ 128 x 128)
//    One block = 16 rows; 8 waves, wave w owns output column tile [16w,16w+16).
//    K loop: 32 iterations of V_WMMA_F32_16X16X4_F32.
//    A (16x4 f32) per ISA 7.12.2: lane L holds row M=L&15,
//      VGPR0/1 = K = 2*(L>>4) + {0,1}. B (4x16) mirrored by column.
//    C/D: VGPR r -> M = r + 8*(L>>4), N = L&15.
// ---------------------------------------------------------------------------
__global__ void han_proj_wmma_kernel(const float* __restrict__ x,
                                     const float* __restrict__ w,
                                     const float* __restrict__ bias,
                                     float* __restrict__ h) {
    __shared__ float Atile[16 * 132];              // padded stride -> no bank conflicts
    const int rowBase = blockIdx.x * 16;
    const int wave = threadIdx.x >> 5;             // 0..7 (column tile)
    const int lane = threadIdx.x & 31;

    for (int i = threadIdx.x; i < 16 * 128; i += blockDim.x) {
        int r = i >> 7, c = i & 127;
        Atile[r * 132 + c] = x[(size_t)(rowBase + r) * DHVAL + c];
    }
    __syncthreads();

    const int colBase = wave * 16;
    const int col  = lane & 15;
    const int arow = lane & 15;
    const int kgrp = lane >> 4;                    // 0 or 1
    v8f acc = {};
    for (int k0 = 0; k0 < 128; k0 += 4) {
        const int ka = k0 + 2 * kgrp;
        v2f a, b;
        a.x = Atile[arow * 132 + ka];
        a.y = Atile[arow * 132 + ka + 1];
        b.x = w[(size_t)ka * DHVAL + colBase + col];
        b.y = w[(size_t)(ka + 1) * DHVAL + colBase + col];
        acc = __builtin_amdgcn_wmma_f32_16x16x4_f32(false, a, false, b,
                                                    (short)0, acc, false, false);
    }
    const float bv = bias[colBase + col];
    const int mBase = kgrp * 8;
#pragma unroll
    for (int r = 0; r < 8; ++r)
        h[(size_t)(rowBase + mBase + r) * DHVAL + colBase + col] = acc[r] + bv;
}

// ---------------------------------------------------------------------------
// 2) Per-node head scores: alpha[n,h] = sum_d h[n,h,d] * a[h,d]  (4 variants)
// ---------------------------------------------------------------------------
__global__ void han_alpha_kernel(const float* __restrict__ h,
                                 const float* __restrict__ ls0, const float* __restrict__ ld0,
                                 const float* __restrict__ ls1, const float* __restrict__ ld1,
                                 float* __restrict__ as0, float* __restrict__ ad0,
                                 float* __restrict__ as1, float* __restrict__ ad1,
                                 int total) {
    int idx = blockIdx.x * blockDim.x + threadIdx.x;
    if (idx >= total) return;
    const int n = idx >> 3, hh = idx & 7;
    const float* hp = h + (size_t)n * DHVAL + hh * DVAL;
    const float* p0 = ls0 + hh * DVAL;  const float* q0 = ld0 + hh * DVAL;
    const float* p1 = ls1 + hh * DVAL;  const float* q1 = ld1 + hh * DVAL;
    float a0 = 0.f, b0 = 0.f, a1 = 0.f, b1 = 0.f;
#pragma unroll
    for (int d = 0; d < DVAL; ++d) {
        const float v = hp[d];
        a0 += v * p0[d]; b0 += v * q0[d];
        a1 += v * p1[d]; b1 += v * q1[d];
    }
    as0[idx] = a0; ad0[idx] = b0; as1[idx] = a1; ad1[idx] = b1;
}

// ---------------------------------------------------------------------------
// 3a) Segment max: m[dst,h] = max over edges of leaky_relu(as[src]+ad[dst])
// ---------------------------------------------------------------------------
__global__ void han_edge_max_kernel(const int* __restrict__ src, const int* __restrict__ dst,
                                    const float* __restrict__ as, const float* __restrict__ ad,
                                    unsigned* __restrict__ m, int ne) {
    int e = blockIdx.x * blockDim.x + threadIdx.x;
    if (e >= ne) return;
    const int s = src[e], d = dst[e];
#pragma unroll
    for (int hh = 0; hh < HVAL; ++hh) {
        float a = as[s * HVAL + hh] + ad[d * HVAL + hh];
        a = (a > 0.f) ? a : NEG_SLOPE * a;
        atomicMax(&m[d * HVAL + hh], encf(a));
    }
}

// ---------------------------------------------------------------------------
// 3b) Segment exp-sum: s[dst,h] += exp(alpha - m[dst,h])
// ---------------------------------------------------------------------------
__global__ void han_edge_sum_kernel(const int* __restrict__ src, const int* __restrict__ dst,
                                    const float* __restrict__ as, const float* __restrict__ ad,
                                    const unsigned* __restrict__ m, float* __restrict__ ssum,
                                    int ne) {
    int e = blockIdx.x * blockDim.x + threadIdx.x;
    if (e >= ne) return;
    const int s = src[e], d = dst[e];
#pragma unroll
    for (int hh = 0; hh < HVAL; ++hh) {
        float a = as[s * HVAL + hh] + ad[d * HVAL + hh];
        a = (a > 0.f) ? a : NEG_SLOPE * a;
        atomicAdd(&ssum[d * HVAL + hh], expf(a - decf(m[d * HVAL + hh])));
    }
}

// ---------------------------------------------------------------------------
// 3c) Weighted scatter: out[dst,h,:] += w_e * h[src,h,:]
//     One thread per (edge, head); 8 consecutive threads share an edge so the
//     512B h[src] row is fetched in coalesced 64B chunks. h and out are both
//     51MB -> L2-resident on the 192MB L2.
// ---------------------------------------------------------------------------
__global__ void han_edge_scatter_kernel(const int* __restrict__ src, const int* __restrict__ dst,
                                        const float* __restrict__ as, const float* __restrict__ ad,
                                        const unsigned* __restrict__ m, const float* __restrict__ ssum,
                                        const float* __restrict__ h, float* __restrict__ out,
                                        int total) {
    int idx = blockIdx.x * blockDim.x + threadIdx.x;
    if (idx >= total) return;
    const int e = idx >> 3, hh = idx & 7;
    const int s = src[e], d = dst[e];
    float a = as[s * HVAL + hh] + ad[d * HVAL + hh];
    a = (a > 0.f) ? a : NEG_SLOPE * a;
    const float w = expf(a - decf(m[d * HVAL + hh])) / (ssum[d * HVAL + hh] + 1e-16f);
    const float* hp = h + (size_t)s * DHVAL + hh * DVAL;
    float* op = out + (size_t)d * DHVAL + hh * DVAL;
    __builtin_prefetch(hp, 0, 0);                  // global_prefetch_b8
#pragma unroll
    for (int dd = 0; dd < DVAL; ++dd)
        atomicAdd(&op[dd], w * hp[dd]);
}

// ---------------------------------------------------------------------------
// 4) Semantic score: score[m] = (1/N) * sum_n q . tanh(relu(out_m[n]) @ k_w + k_b)
//    One block per (node, metapath); thread d owns output column d.
// ---------------------------------------------------------------------------
__global__ void han_semantic_score_kernel(const float* __restrict__ out0,
                                          const float* __restrict__ out1,
                                          const float* __restrict__ kw,
                                          const float* __restrict__ kb,
                                          const float* __restrict__ q,
                                          float* __restrict__ score, int n) {
    __shared__ float row[DHVAL];
    __shared__ float red[DHVAL];
    const int m = blockIdx.y;
    const int node = blockIdx.x;
    const int d = threadIdx.x;
    const float* o = (m == 0 ? out0 : out1) + (size_t)node * DHVAL;
    float v = o[d];
    row[d] = (v > 0.f) ? v : 0.f;
    __syncthreads();
    float acc = kb[d];
    for (int k = 0; k < DHVAL; ++k)
        acc += row[k] * kw[(size_t)k * DHVAL + d];
    red[d] = tanhf(acc) * q[d];
    __syncthreads();
    for (int st = 64; st > 0; st >>= 1) {
        if (d < st) red[d] += red[d + st];
        __syncthreads();
    }
    if (d == 0) atomicAdd(&score[m], red[0] / (float)n);
}

// ---------------------------------------------------------------------------
// 5) beta = softmax(score)  (M=2)
// ---------------------------------------------------------------------------
__global__ void han_beta_kernel(const float* __restrict__ score, float* __restrict__ beta) {
    if (threadIdx.x == 0) {
        const float a = score[0], b = score[1];
        const float mx = fmaxf(a, b);
        const float ea = expf(a - mx), eb = expf(b - mx);
        const float s = ea + eb;
        beta[0] = ea / s; beta[1] = eb / s;
    }
}

// ---------------------------------------------------------------------------
// 6) Fused output: y = (beta0*relu(out0)+beta1*relu(out1)) @ out_w + out_b
//    Same WMMA tiling as the projection; DOUT=64 -> 4 column waves.
// ---------------------------------------------------------------------------
__global__ void han_fuse_out_wmma_kernel(const float* __restrict__ out0,
                                         const float* __restrict__ out1,
                                         const float* __restrict__ beta,
                                         const float* __restrict__ ow,
                                         const float* __restrict__ ob,
                                         float* __restrict__ y) {
    __shared__ float Atile[16 * 132];
    const int rowBase = blockIdx.x * 16;
    const int wave = threadIdx.x >> 5;             // 0..3
    const int lane = threadIdx.x & 31;
    const float b0 = beta[0], b1 = beta[1];

    for (int i = threadIdx.x; i < 16 * 128; i += blockDim.x) {
        int r = i >> 7, c = i & 127;
        float v0 = out0[(size_t)(rowBase + r) * DHVAL + c]; v0 = (v0 > 0.f) ? v0 : 0.f;
        float v1 = out1[(size_t)(rowBase + r) * DHVAL + c]; v1 = (v1 > 0.f) ? v1 : 0.f;
        Atile[r * 132 + c] = b0 * v0 + b1 * v1;
    }
    __syncthreads();

    const int colBase = wave * 16;
    const int col  = lane & 15;
    const int arow = lane & 15;
    const int kgrp = lane >> 4;
    v8f acc = {};
    for (int k0 = 0; k0 < 128; k0 += 4) {
        const int ka = k0 + 2 * kgrp;
        v2f a, b;
        a.x = Atile[arow * 132 + ka];
        a.y = Atile[arow * 132 + ka + 1];
        b.x = ow[(size_t)ka * DOUTVAL + colBase + col];
        b.y = ow[(size_t)(ka + 1) * DOUTVAL + colBase + col];
        acc = __builtin_amdgcn_wmma_f32_16x16x4_f32(false, a, false, b,
                                                    (short)0, acc, false, false);
    }
    const float bv = ob[colBase + col];
    const int mBase = kgrp * 8;
#pragma unroll
    for (int r = 0; r < 8; ++r)
        y[(size_t)(rowBase + mBase + r) * DOUTVAL + colBase + col] = acc[r] + bv;
}

// ---------------------------------------------------------------------------
extern "C" void kernel_launch(void* const* d_in, const int* in_sizes, int n_in,
                              void* d_out, int out_size, void* d_ws, size_t ws_size,
                              hipStream_t stream) {
    const float* x      = (const float*)d_in[0];
    const int*   ei0    = (const int*)  d_in[1];
    const int*   ei1    = (const int*)  d_in[2];
    const float* proj_w = (const float*)d_in[3];
    const float* proj_b = (const float*)d_in[4];
    const float* ls0    = (const float*)d_in[5];
    const float* ld0    = (const float*)d_in[6];
    const float* ls1    = (const float*)d_in[7];
    const float* ld1    = (const float*)d_in[8];
    const float* kw     = (const float*)d_in[9];
    const float* kb     = (const float*)d_in[10];
    const float* q      = (const float*)d_in[11];
    const float* ow     = (const float*)d_in[12];
    const float* ob     = (const float*)d_in[13];

    const int Nn = in_sizes[0] / DHVAL;            // 100000 (DIN==128)
    const int Ee = in_sizes[1] / 2;                // 1600000
    const int nh  = Nn * HVAL;
    const int ndh = Nn * DHVAL;

    // Workspace carve-up (floats)
    float*    ws   = (float*)d_ws;
    float*    h    = ws;                               // N*128
    float*    as0  = h   + (size_t)Nn * DHVAL;         // N*8 each
    float*    ad0  = as0 + nh;
    float*    as1  = ad0 + nh;
    float*    ad1  = as1 + nh;
    unsigned* m0   = (unsigned*)(ad1 + nh);            // N*8 each
    unsigned* m1   = m0 + nh;
    float*    s0   = (float*)(m1 + nh);                // N*8 each
    float*    s1   = s0 + nh;
    float*    out0 = s1 + nh;                          // N*128 each
    float*    out1 = out0 + (size_t)Nn * DHVAL;
    float*    score = out1 + (size_t)Nn * DHVAL;       // 2
    float*    beta  = score + 2;                       // 2

    const int THR = 256;

    han_init_kernel<<<(ndh + THR - 1) / THR, THR, 0, stream>>>(
        m0, m1, s0, s1, out0, out1, score, nh, ndh);

    han_proj_wmma_kernel<<<Nn / 16, 256, 0, stream>>>(x, proj_w, proj_b, h);

    han_alpha_kernel<<<(nh + THR - 1) / THR, THR, 0, stream>>>(
        h, ls0, ld0, ls1, ld1, as0, ad0, as1, ad1, nh);

    // --- metapath 0 ---
    han_edge_max_kernel<<<(Ee + THR - 1) / THR, THR, 0, stream>>>(
        ei0, ei0 + Ee, as0, ad0, m0, Ee);
    han_edge_sum_kernel<<<(Ee + THR - 1) / THR, THR, 0, stream>>>(
        ei0, ei0 + Ee, as0, ad0, m0, s0, Ee);
    han_edge_scatter_kernel<<<(Ee * HVAL + THR - 1) / THR, THR, 0, stream>>>(
        ei0, ei0 + Ee, as0, ad0, m0, s0, h, out0, Ee * HVAL);

    // --- metapath 1 ---
    han_edge_max_kernel<<<(Ee + THR - 1) / THR, THR, 0, stream>>>(
        ei1, ei1 + Ee, as1, ad1, m1, Ee);
    han_edge_sum_kernel<<<(Ee + THR - 1) / THR, THR, 0, stream>>>(
        ei1, ei1 + Ee, as1, ad1, m1, s1, Ee);
    han_edge_scatter_kernel<<<(Ee * HVAL + THR - 1) / THR, THR, 0, stream>>>(
        ei1, ei1 + Ee, as1, ad1, m1, s1, h, out1, Ee * HVAL);

    // --- semantic attention + fused output projection ---
    dim3 sgrid(Nn, 2);
    han_semantic_score_kernel<<<sgrid, DHVAL, 0, stream>>>(
        out0, out1, kw, kb, q, score, Nn);
    han_beta_kernel<<<1, 32, 0, stream>>>(score, beta);
    han_fuse_out_wmma_kernel<<<Nn / 16, 128, 0, stream>>>(
        out0, out1, beta, ow, ob, (float*)d_out);
}